// LocalAttentionBlock_27401891349144
// MI455X (gfx1250) — compile-verified
//
#include <hip/hip_runtime.h>
#include <math.h>

typedef __attribute__((ext_vector_type(2))) float v2f;
typedef __attribute__((ext_vector_type(8))) float v8f;

#define WIDTH_  1024
#define NHEADS_ 16
#define HDIM_   64
#define WINSZ_  256
#define TLEN_   2048
#define BATCH_  2

#define KP_   16      // K-panel depth per LDS stage
#define KPP_  20      // padded LDS row stride in floats (80B: 16B-aligned, bank-conflict-free)

// D = A(16x4) * B(4x16) + C, full f32 WMMA (CDNA5)
__device__ __forceinline__ v8f wmma4(v2f a, v2f b, v8f c) {
  return __builtin_amdgcn_wmma_f32_16x16x4_f32(
      /*neg_a=*/false, a, /*neg_b=*/false, b,
      /*c_mod=*/(short)0, c, /*reuse_a=*/false, /*reuse_b=*/false);
}

// Async 16B/lane copy global -> LDS (gfx1250 ASYNCcnt path).
__device__ __forceinline__ void async_cp16(unsigned int lds_off, const void* gptr) {
  asm volatile("global_load_async_to_lds_b128 %0, %1, off"
               :: "v"(lds_off), "v"(gptr) : "memory");
}

// ---------------------------------------------------------------------------
// LDS-tiled C[M,N] = A[M,K] @ W[N,K]^T (+bias). Block tile 128x128, 8 waves
// of 32(M)x64(N); K-panels of 16 double-buffered in LDS via async-to-LDS.
// Requires M%128==0, N%128==0, K%16==0.
// ---------------------------------------------------------------------------
__global__ __launch_bounds__(256) void gemm_xwt_tiled_kernel(
    const float* __restrict__ A, const float* __restrict__ W,
    const float* __restrict__ bias, float* __restrict__ C,
    int M, int N, int K)
{
  __shared__ __align__(16) float Asm[2][128 * KPP_];
  __shared__ __align__(16) float Wsm[2][128 * KPP_];

  const int tid  = threadIdx.x;
  const int wave = tid >> 5;        // 0..7
  const int lane = tid & 31;
  const int ln   = lane & 15;
  const int hi   = lane >> 4;
  const int wm   = wave >> 1;       // 0..3 : 32-row slice
  const int wn   = wave & 1;        // 0..1 : 64-col slice

  const int gm0 = blockIdx.y * 128;
  const int gn0 = blockIdx.x * 128;

  const int nk = K / KP_;

  // Per stage: each panel is 128 rows x 16 floats = 512 float4; 2 per thread.
  auto issue = [&](int kp) {
    const int buf = kp & 1;
    const float* ag = A + (size_t)gm0 * K + (size_t)kp * KP_;
    const float* wg = W + (size_t)gn0 * K + (size_t)kp * KP_;
#pragma unroll
    for (int j = 0; j < 2; ++j) {
      const int idx = tid + j * 256;        // 0..511
      const int row = idx >> 2;             // 0..127
      const int c4  = idx & 3;              // float4 column
      async_cp16((unsigned int)(uintptr_t)&Asm[buf][row * KPP_ + c4 * 4],
                 ag + (size_t)row * K + c4 * 4);
      async_cp16((unsigned int)(uintptr_t)&Wsm[buf][row * KPP_ + c4 * 4],
                 wg + (size_t)row * K + c4 * 4);
    }
  };

  v8f acc[2][4] = {};

  issue(0);
  if (nk > 1) issue(1);

  for (int kp = 0; kp < nk; ++kp) {
    // Wait for this stage's 4 async ops (the next stage's 4 may stay in flight).
    if (kp + 1 < nk) asm volatile("s_wait_asynccnt 0x4" ::: "memory");
    else             asm volatile("s_wait_asynccnt 0x0" ::: "memory");
    __syncthreads();

    const float* As = &Asm[kp & 1][0];
    const float* Ws = &Wsm[kp & 1][0];
#pragma unroll
    for (int kk = 0; kk < KP_; kk += 4) {
      v2f af[2], wf[4];
#pragma unroll
      for (int mi = 0; mi < 2; ++mi)
        af[mi] = *(const v2f*)(As + (wm * 32 + mi * 16 + ln) * KPP_ + kk + 2 * hi);
#pragma unroll
      for (int ni = 0; ni < 4; ++ni)
        wf[ni] = *(const v2f*)(Ws + (wn * 64 + ni * 16 + ln) * KPP_ + kk + 2 * hi);
#pragma unroll
      for (int mi = 0; mi < 2; ++mi)
#pragma unroll
        for (int ni = 0; ni < 4; ++ni)
          acc[mi][ni] = wmma4(af[mi], wf[ni], acc[mi][ni]);
    }
    __syncthreads();
    if (kp + 2 < nk) issue(kp + 2);   // safe: all waves past compute barrier
  }

#pragma unroll
  for (int mi = 0; mi < 2; ++mi) {
#pragma unroll
    for (int ni = 0; ni < 4; ++ni) {
      const int n = gn0 + wn * 64 + ni * 16 + ln;
      const float bv = bias ? bias[n] : 0.0f;
#pragma unroll
      for (int i = 0; i < 8; ++i) {
        const int m = gm0 + wm * 32 + mi * 16 + i + 8 * hi;
        C[(size_t)m * N + n] = acc[mi][ni][i] + bv;
      }
    }
  }
}

// ---------------------------------------------------------------------------
// Direct-WMMA GEMM for the small K/V projections (N=64).
// ---------------------------------------------------------------------------
__global__ __launch_bounds__(128) void gemm_xwt_kernel(
    const float* __restrict__ A, const float* __restrict__ W,
    const float* __restrict__ bias, float* __restrict__ C,
    int M, int N, int K)
{
  const int tid  = threadIdx.x;
  const int wave = tid >> 5;
  const int lane = tid & 31;
  const int ln   = lane & 15;
  const int hi   = lane >> 4;

  const int m0 = (blockIdx.y * 4 + wave) * 16;
  const int n0 = blockIdx.x * 64;
  if (m0 >= M) return;

  v8f acc[4] = {};
  const float* arow = A + (size_t)(m0 + ln) * K + 2 * hi;
  const float* wrow = W + (size_t)(n0 + ln) * K + 2 * hi;
  const size_t ws16 = (size_t)16 * K;

  for (int kk = 0; kk < K; kk += 4) {
    v2f a = *(const v2f*)(arow + kk);
    v2f b0 = *(const v2f*)(wrow + kk);
    v2f b1 = *(const v2f*)(wrow + ws16 + kk);
    v2f b2 = *(const v2f*)(wrow + 2 * ws16 + kk);
    v2f b3 = *(const v2f*)(wrow + 3 * ws16 + kk);
    acc[0] = wmma4(a, b0, acc[0]);
    acc[1] = wmma4(a, b1, acc[1]);
    acc[2] = wmma4(a, b2, acc[2]);
    acc[3] = wmma4(a, b3, acc[3]);
  }

#pragma unroll
  for (int t = 0; t < 4; ++t) {
    const int n = n0 + 16 * t + ln;
    const float bv = bias ? bias[n] : 0.0f;
#pragma unroll
    for (int i = 0; i < 8; ++i) {
      const int m = m0 + i + 8 * hi;
      C[(size_t)m * N + n] = acc[t][i] + bv;
    }
  }
}

// ---------------------------------------------------------------------------
// Windowed (causal, window=256) attention, one shared KV head.
// One wave per (batch, head, 16-query tile); flash-style online softmax over
// 17 key tiles; loads are batched ahead of each WMMA chain.
// ---------------------------------------------------------------------------
__global__ __launch_bounds__(128) void local_attn_kernel(
    const float* __restrict__ Qg, const float* __restrict__ Kg,
    const float* __restrict__ Vg, float* __restrict__ Og)
{
  __shared__ float pbuf[4][16][17];   // per-wave P transpose staging (padded)

  const int tid  = threadIdx.x;
  const int wave = tid >> 5;
  const int lane = tid & 31;
  const int ln   = lane & 15;
  const int hi   = lane >> 4;

  const int qt   = blockIdx.x * 4 + wave;   // 0..127
  const int q0   = qt * 16;
  const int head = blockIdx.y;
  const int bb   = blockIdx.z;

  const float* Q = Qg + (size_t)bb * TLEN_ * WIDTH_ + head * HDIM_;
  const float* K = Kg + (size_t)bb * TLEN_ * HDIM_;
  const float* V = Vg + (size_t)bb * TLEN_ * HDIM_;

  // Preload all 16 Q A-fragments (16 queries x 64 dims).
  v2f qf[16];
#pragma unroll
  for (int s = 0; s < 16; ++s)
    qf[s] = *(const v2f*)(Q + (size_t)(q0 + ln) * WIDTH_ + s * 4 + 2 * hi);

  v8f acc[4] = {};
  float mrow[8], lrow[8];
#pragma unroll
  for (int i = 0; i < 8; ++i) { mrow[i] = -__builtin_inff(); lrow[i] = 0.0f; }

  const float scale = 0.125f;   // 1/sqrt(64)
  const int s_lo = (q0 > WINSZ_) ? (q0 - WINSZ_) : 0;

  for (int s0 = s_lo; s0 <= q0; s0 += 16) {
    if (s0 + 16 <= q0) {  // prefetch next key/value tile rows
      __builtin_prefetch(K + (size_t)(s0 + 16 + ln) * HDIM_, 0, 3);
      __builtin_prefetch(V + (size_t)(s0 + 16 + ln) * HDIM_, 0, 3);
    }

    // --- logits: batch all 16 B-fragment loads, then 16 back-to-back WMMAs
    v2f kf[16];
#pragma unroll
    for (int s = 0; s < 16; ++s)
      kf[s] = *(const v2f*)(K + (size_t)(s0 + ln) * HDIM_ + s * 4 + 2 * hi);
    v8f L = {};
#pragma unroll
    for (int s = 0; s < 16; ++s)
      L = wmma4(qf[s], kf[s], L);

    // --- mask + online softmax (row M = i + 8*hi lives in one 16-lane half)
    const int sn = s0 + ln;
    float fac[8], prob[8];
#pragma unroll
    for (int i = 0; i < 8; ++i) {
      const int qm = q0 + i + 8 * hi;
      const bool valid = (qm >= sn) && (qm <= sn + WINSZ_);
      const float lv = valid ? L[i] * scale : -1.0e30f;

      float rm = lv;
      rm = fmaxf(rm, __shfl_xor(rm, 1, 16));
      rm = fmaxf(rm, __shfl_xor(rm, 2, 16));
      rm = fmaxf(rm, __shfl_xor(rm, 4, 16));
      rm = fmaxf(rm, __shfl_xor(rm, 8, 16));
      const float mnew = fmaxf(mrow[i], rm);

      const float p = __expf(lv - mnew);
      float rs = p;
      rs += __shfl_xor(rs, 1, 16);
      rs += __shfl_xor(rs, 2, 16);
      rs += __shfl_xor(rs, 4, 16);
      rs += __shfl_xor(rs, 8, 16);

      fac[i]  = __expf(mrow[i] - mnew);
      lrow[i] = lrow[i] * fac[i] + rs;
      mrow[i] = mnew;
      prob[i] = p;
    }

#pragma unroll
    for (int t = 0; t < 4; ++t)
#pragma unroll
      for (int i = 0; i < 8; ++i)
        acc[t][i] *= fac[i];

    // --- P: C-layout -> A-layout via this wave's private LDS slice
#pragma unroll
    for (int i = 0; i < 8; ++i)
      pbuf[wave][i + 8 * hi][ln] = prob[i];

    // --- batch all V B-fragment loads, then the 16-WMMA PV chain
    v2f vf[4][4];
#pragma unroll
    for (int k4 = 0; k4 < 4; ++k4)
#pragma unroll
      for (int t = 0; t < 4; ++t) {
        vf[k4][t].x = V[(size_t)(s0 + k4 * 4 + 2 * hi)     * HDIM_ + t * 16 + ln];
        vf[k4][t].y = V[(size_t)(s0 + k4 * 4 + 2 * hi + 1) * HDIM_ + t * 16 + ln];
      }
#pragma unroll
    for (int k4 = 0; k4 < 4; ++k4) {
      v2f ap;
      ap.x = pbuf[wave][ln][k4 * 4 + 2 * hi];
      ap.y = pbuf[wave][ln][k4 * 4 + 2 * hi + 1];
#pragma unroll
      for (int t = 0; t < 4; ++t)
        acc[t] = wmma4(ap, vf[k4][t], acc[t]);
    }
  }

  // Normalize and store (b, t, heads*64).
#pragma unroll
  for (int i = 0; i < 8; ++i) {
    const float inv = 1.0f / lrow[i];
    const int m = q0 + i + 8 * hi;
#pragma unroll
    for (int t = 0; t < 4; ++t)
      Og[((size_t)bb * TLEN_ + m) * WIDTH_ + head * HDIM_ + t * 16 + ln] =
          acc[t][i] * inv;
  }
}

extern "C" void kernel_launch(void* const* d_in, const int* in_sizes, int n_in,
                              void* d_out, int out_size, void* d_ws, size_t ws_size,
                              hipStream_t stream) {
  (void)in_sizes; (void)n_in; (void)out_size; (void)ws_size;
  const float* x  = (const float*)d_in[0];
  // d_in[1]: segment_pos == arange; mask uses indices directly
  const float* Wq = (const float*)d_in[2];
  const float* Wk = (const float*)d_in[3];
  const float* Wv = (const float*)d_in[4];
  const float* Wf = (const float*)d_in[5];
  const float* bf = (const float*)d_in[6];
  float* out = (float*)d_out;

  const int M = BATCH_ * TLEN_;   // 4096 rows

  float* qbuf = (float*)d_ws;                       // M * 1024
  float* kbuf = qbuf + (size_t)M * WIDTH_;          // M * 64
  float* vbuf = kbuf + (size_t)M * HDIM_;           // M * 64
  float* abuf = vbuf + (size_t)M * HDIM_;           // M * 1024

  // Q projection: LDS-tiled async GEMM
  gemm_xwt_tiled_kernel<<<dim3(WIDTH_ / 128, M / 128), dim3(256), 0, stream>>>(
      x, Wq, nullptr, qbuf, M, WIDTH_, WIDTH_);
  // K/V projections (N=64): direct-WMMA GEMM
  gemm_xwt_kernel<<<dim3(1, M / 64), dim3(128), 0, stream>>>(
      x, Wk, nullptr, kbuf, M, HDIM_, WIDTH_);
  gemm_xwt_kernel<<<dim3(1, M / 64), dim3(128), 0, stream>>>(
      x, Wv, nullptr, vbuf, M, HDIM_, WIDTH_);

  // Windowed attention: one wave per (b, head, 16-query tile)
  local_attn_kernel<<<dim3(TLEN_ / 64, NHEADS_, BATCH_), dim3(128), 0, stream>>>(
      qbuf, kbuf, vbuf, abuf);

  // Output projection with bias: LDS-tiled async GEMM
  gemm_xwt_tiled_kernel<<<dim3(WIDTH_ / 128, M / 128), dim3(256), 0, stream>>>(
      abuf, Wf, bf, out, M, WIDTH_, WIDTH_);
}